// MoERouterV2_4595615007350
// MI455X (gfx1250) — compile-verified
//
#include <hip/hip_runtime.h>
#include <math.h>

// MoE router for MI455X (gfx1250, wave32).
//
// Roofline: GEMM = 8.6 GFLOP over 134 MB of fp32 activations ->
// ~64 FLOP/byte, HBM floor ~5.8us @ 23.3 TB/s. Router math is fp32 by
// definition, so we use V_WMMA_F32_16X16X4_F32 (the only f32 WMMA shape),
// which keeps compute near the memory floor. The 1 MB router weight is
// re-read by every wave but stays resident in the 192 MB L2.

#define D_MODEL        2048
#define NUM_EXPERTS    128
#define TOP_K          8
#define WAVES_PER_BLOCK 4
#define TOKENS_PER_WAVE 16
#define LDS_STRIDE     132   // 128 + 4 pad: conflict-free row-major scans

typedef float v2f __attribute__((ext_vector_type(2)));
typedef float v8f __attribute__((ext_vector_type(8)));

__global__ __launch_bounds__(WAVES_PER_BLOCK * 32)
void moe_router_wmma(const float* __restrict__ x,
                     const float* __restrict__ weight,
                     const float* __restrict__ score_bias,
                     float* __restrict__ out,
                     int n_tokens) {
  __shared__ float lds[WAVES_PER_BLOCK][TOKENS_PER_WAVE][LDS_STRIDE];

  const int lane = threadIdx.x & 31;
  const int wave = threadIdx.x >> 5;
  const int half = lane >> 4;   // hi/lo 16-lane group
  const int r    = lane & 15;

  const int tokenBase = (blockIdx.x * WAVES_PER_BLOCK + wave) * TOKENS_PER_WAVE;
  if (tokenBase >= n_tokens) return;  // wave-uniform; EXEC stays all-ones

  // ---------------- GEMM: logits[16 tok][128 exp] ----------------
  // A fragment (16x4 f32): lanes 0-15 hold (k0,k1) of row M=r,
  //                        lanes 16-31 hold (k2,k3) of row M=r.
  // B fragment (4x16 f32): lanes 0-15 hold (k0,k1) of col N=r,
  //                        lanes 16-31 hold (k2,k3) of col N=r.
  // The logical k->global-column map is ours to choose as long as A and B
  // agree, so one b128 (8 columns across the two half-wave groups) feeds
  // two WMMAs: cols {kb,kb+1 | kb+4,kb+5} then {kb+2,kb+3 | kb+6,kb+7}.
  const float* xA = x + (size_t)(tokenBase + r) * D_MODEL + 4 * half;
  const float* wB = weight + (size_t)r * D_MODEL + 4 * half;

  v8f acc[8];
#pragma unroll
  for (int t = 0; t < 8; ++t) acc[t] = (v8f){0, 0, 0, 0, 0, 0, 0, 0};

  for (int kb = 0; kb < D_MODEL; kb += 8) {
    const float4 a4 = *(const float4*)(xA + kb);
    const v2f a0 = {a4.x, a4.y};
    const v2f a1 = {a4.z, a4.w};
#pragma unroll
    for (int t = 0; t < 8; ++t) {
      const float4 b4 = *(const float4*)(wB + (size_t)t * 16 * D_MODEL + kb);
      const v2f b0 = {b4.x, b4.y};
      const v2f b1 = {b4.z, b4.w};
      acc[t] = __builtin_amdgcn_wmma_f32_16x16x4_f32(
          false, a0, false, b0, (short)0, acc[t], false, false);
      acc[t] = __builtin_amdgcn_wmma_f32_16x16x4_f32(
          false, a1, false, b1, (short)0, acc[t], false, false);
    }
  }

  // C/D layout: VGPR i, lanes 0-15 -> M=i, lanes 16-31 -> M=i+8; N = r.
#pragma unroll
  for (int t = 0; t < 8; ++t) {
#pragma unroll
    for (int i = 0; i < 8; ++i) {
      lds[wave][i + 8 * half][t * 16 + r] = acc[t][i];
    }
  }
  __syncthreads();

  // ---------------- softmax + biased top-8 + L2 norm ----------------
  if (lane < TOKENS_PER_WAVE) {
    float* row = lds[wave][lane];

    float mx = -INFINITY;
    for (int e = 0; e < NUM_EXPERTS; ++e) mx = fmaxf(mx, row[e]);

    float s = 0.0f;
    for (int e = 0; e < NUM_EXPERTS; ++e) {
      const float p = __expf(row[e] - mx);
      row[e] = p;
      s += p;
    }
    const float invs = 1.0f / s;

    // Descending top-8 on (prob + bias); strict '>' keeps earlier index on
    // ties, matching jax.lax.top_k.
    float bv[TOP_K], bp[TOP_K];
    int bi[TOP_K];
#pragma unroll
    for (int j = 0; j < TOP_K; ++j) { bv[j] = -INFINITY; bp[j] = 0.0f; bi[j] = 0; }

    for (int e = 0; e < NUM_EXPERTS; ++e) {
      const float p = row[e] * invs;
      float cv = p + score_bias[e];
      float cp = p;
      int   ci = e;
#pragma unroll
      for (int j = 0; j < TOP_K; ++j) {
        const bool ins = cv > bv[j];
        const float tv = bv[j], tp = bp[j];
        const int   ti = bi[j];
        if (ins) {
          bv[j] = cv; bp[j] = cp; bi[j] = ci;
          cv = tv; cp = tp; ci = ti;
        }
      }
    }

    float s2 = 0.0f;
#pragma unroll
    for (int j = 0; j < TOP_K; ++j) s2 += bp[j] * bp[j];
    const float invn = 1.0f / sqrtf(s2);

    const size_t tok = (size_t)tokenBase + lane;
    float* out_w = out;
    float* out_i = out + (size_t)n_tokens * TOP_K;
#pragma unroll
    for (int j = 0; j < TOP_K; ++j) {
      out_w[tok * TOP_K + j] = bp[j] * invn;
      out_i[tok * TOP_K + j] = (float)bi[j];
    }
  }
}

extern "C" void kernel_launch(void* const* d_in, const int* in_sizes, int n_in,
                              void* d_out, int out_size, void* d_ws, size_t ws_size,
                              hipStream_t stream) {
  const float* x    = (const float*)d_in[0];
  const float* wgt  = (const float*)d_in[1];
  const float* bias = (const float*)d_in[2];

  const int n_tokens = in_sizes[0] / D_MODEL;  // 16384 for the reference shapes
  const int tokens_per_block = WAVES_PER_BLOCK * TOKENS_PER_WAVE;
  const int grid = (n_tokens + tokens_per_block - 1) / tokens_per_block;

  moe_router_wmma<<<grid, WAVES_PER_BLOCK * 32, 0, stream>>>(
      x, wgt, bias, (float*)d_out, n_tokens);
}